// GNNStructEncoder_206158430347
// MI455X (gfx1250) — compile-verified
//
#include <hip/hip_runtime.h>
#include <hip/hip_bf16.h>

// Problem constants (match reference)
#define NN    50000
#define EE    800000
#define INF_  128
#define HH    96
#define RTILES 3125        // NN / 16
#define CTILES 6           // HH / 16

typedef float v2f __attribute__((ext_vector_type(2)));
typedef float v8f __attribute__((ext_vector_type(8)));

// ---------------------------------------------------------------- zero scratch
__global__ void k_zero(float* __restrict__ p, int n) {
    int i = blockIdx.x * blockDim.x + threadIdx.x;
    if (i < n) p[i] = 0.0f;
}

// ---------------------------------------------------------------- GEMM0: h0 = x @ W0^T + b0
// one wave per 16-row tile; 6 column tiles (96 cols) accumulated in-register
__global__ void k_gemm0(const float* __restrict__ x, const float* __restrict__ W0,
                        const float* __restrict__ b0, float* __restrict__ h0) {
    int wave = (blockIdx.x * blockDim.x + threadIdx.x) >> 5;
    if (wave >= RTILES) return;
    int lane = threadIdx.x & 31;
    int half = lane >> 4;           // 0: K=+0,+1   1: K=+2,+3
    int l16  = lane & 15;
    int row  = wave * 16 + l16;

    v8f acc[CTILES] = {};
    const float* arow = x + (size_t)row * INF_ + 2 * half;

    for (int k = 0; k < INF_; k += 4) {
        v2f a = *(const v2f*)(arow + k);
#pragma unroll
        for (int t = 0; t < CTILES; ++t) {
            // B[k'][n] = W0[n][k'] ; n = t*16 + l16 ; k' = k + 2*half (+1)
            v2f b = *(const v2f*)(W0 + (size_t)(t * 16 + l16) * INF_ + k + 2 * half);
            acc[t] = __builtin_amdgcn_wmma_f32_16x16x4_f32(
                false, a, false, b, (short)0, acc[t], false, false);
        }
    }
#pragma unroll
    for (int t = 0; t < CTILES; ++t) {
        int col   = t * 16 + l16;
        float bia = b0[col];
#pragma unroll
        for (int r = 0; r < 8; ++r) {
            int orow = wave * 16 + r + 8 * half;   // C layout: vgpr r -> M=r (lo half) / r+8 (hi half)
            h0[(size_t)orow * HH + col] = acc[t][r] + bia;
        }
    }
}

// ---------------------------------------------------------------- edge scatter: agg[dst] += h0[src]
__global__ void k_scatter(const float* __restrict__ h0, const int* __restrict__ ei,
                          float* __restrict__ agg) {
    int tid = blockIdx.x * blockDim.x + threadIdx.x;
    if (tid >= EE * HH) return;
    int e = tid / HH;
    int f = tid - e * HH;
    int src = ei[e];
    int dst = ei[EE + e];
    atomicAdd(&agg[(size_t)dst * HH + f], h0[(size_t)src * HH + f]);
}

// ---------------------------------------------------------------- GEMM1: h = (h0+agg) @ W1^T + b1
// hpre may alias agg: each wave reads only its own 16 rows, all reads precede stores.
__global__ void k_gemm1(const float* __restrict__ h0, const float* agg,
                        const float* __restrict__ W1, const float* __restrict__ b1,
                        float* hpre) {
    int wave = (blockIdx.x * blockDim.x + threadIdx.x) >> 5;
    if (wave >= RTILES) return;
    int lane = threadIdx.x & 31;
    int half = lane >> 4;
    int l16  = lane & 15;
    int row  = wave * 16 + l16;

    v8f acc[CTILES] = {};
    const float* h0r = h0  + (size_t)row * HH + 2 * half;
    const float* agr = agg + (size_t)row * HH + 2 * half;

    for (int k = 0; k < HH; k += 4) {
        v2f a = *(const v2f*)(h0r + k);
        v2f g = *(const v2f*)(agr + k);
        a = a + g;                                   // z = h0 + agg fused into A fetch
#pragma unroll
        for (int t = 0; t < CTILES; ++t) {
            v2f b = *(const v2f*)(W1 + (size_t)(t * 16 + l16) * HH + k + 2 * half);
            acc[t] = __builtin_amdgcn_wmma_f32_16x16x4_f32(
                false, a, false, b, (short)0, acc[t], false, false);
        }
    }
#pragma unroll
    for (int t = 0; t < CTILES; ++t) {
        int col   = t * 16 + l16;
        float bia = b1[col];
#pragma unroll
        for (int r = 0; r < 8; ++r) {
            int orow = wave * 16 + r + 8 * half;
            hpre[(size_t)orow * HH + col] = acc[t][r] + bia;
        }
    }
}

// ---------------------------------------------------------------- column sum+sumsq (BN stats)
__global__ void k_colstats(const float* __restrict__ h, float* __restrict__ csum,
                           float* __restrict__ csum2) {
    int f = threadIdx.x;                   // 0..95
    float s = 0.0f, s2 = 0.0f;
    for (int i = blockIdx.x; i < NN; i += gridDim.x) {
        float v = h[(size_t)i * HH + f];
        s += v; s2 += v * v;
    }
    atomicAdd(&csum[f], s);
    atomicAdd(&csum2[f], s2);
}

// ---------------------------------------------------------------- fold BN into scale/shift
__global__ void k_bnparams(const float* __restrict__ csum, const float* __restrict__ csum2,
                           const float* __restrict__ gamma, const float* __restrict__ beta,
                           float* __restrict__ scale, float* __restrict__ shift) {
    int f = threadIdx.x;
    const float invN = 1.0f / (float)NN;
    float mean = csum[f] * invN;
    float var  = csum2[f] * invN - mean * mean;     // biased, like torch BN
    float sc   = gamma[f] * rsqrtf(var + 1e-5f);
    scale[f] = sc;
    shift[f] = beta[f] - mean * sc;
}

// ---------------------------------------------------------------- GEMM2: l1 = relu(bn(h)) @ W2^T + b2
__global__ void k_gemm2(const float* __restrict__ hpre, const float* __restrict__ scale,
                        const float* __restrict__ shift, const float* __restrict__ W2,
                        const float* __restrict__ b2, float* __restrict__ l1) {
    int wave = (blockIdx.x * blockDim.x + threadIdx.x) >> 5;
    if (wave >= RTILES) return;
    int lane = threadIdx.x & 31;
    int half = lane >> 4;
    int l16  = lane & 15;
    int row  = wave * 16 + l16;

    v8f acc[CTILES] = {};
    const float* hr = hpre + (size_t)row * HH + 2 * half;

    for (int k = 0; k < HH; k += 4) {
        int kk = k + 2 * half;
        v2f a  = *(const v2f*)(hr + k);
        v2f sc = *(const v2f*)(scale + kk);
        v2f sh = *(const v2f*)(shift + kk);
        a = a * sc + sh;                             // BatchNorm
        a.x = fmaxf(a.x, 0.0f);                      // ReLU
        a.y = fmaxf(a.y, 0.0f);
#pragma unroll
        for (int t = 0; t < CTILES; ++t) {
            v2f b = *(const v2f*)(W2 + (size_t)(t * 16 + l16) * HH + kk);
            acc[t] = __builtin_amdgcn_wmma_f32_16x16x4_f32(
                false, a, false, b, (short)0, acc[t], false, false);
        }
    }
#pragma unroll
    for (int t = 0; t < CTILES; ++t) {
        int col   = t * 16 + l16;
        float bia = b2[col];
#pragma unroll
        for (int r = 0; r < 8; ++r) {
            int orow = wave * 16 + r + 8 * half;
            l1[(size_t)orow * HH + col] = acc[t][r] + bia;
        }
    }
}

// ---------------------------------------------------------------- column sum of l1 (PairNorm mean)
__global__ void k_colsum(const float* __restrict__ l1, float* __restrict__ csum) {
    int f = threadIdx.x;
    float s = 0.0f;
    for (int i = blockIdx.x; i < NN; i += gridDim.x)
        s += l1[(size_t)i * HH + f];
    atomicAdd(&csum[f], s);
}

// ---------------------------------------------------------------- PairNorm 'PN-SCS' in-place
__global__ void k_pairnorm(float* __restrict__ l1, const float* __restrict__ csum) {
    int wid = (blockIdx.x * blockDim.x + threadIdx.x) >> 5;   // one wave per row
    if (wid >= NN) return;
    int lane = threadIdx.x & 31;
    size_t base = (size_t)wid * HH;
    float v0 = l1[base + lane];
    float v1 = l1[base + lane + 32];
    float v2 = l1[base + lane + 64];
    float s = v0 * v0 + v1 * v1 + v2 * v2;
#pragma unroll
    for (int off = 16; off > 0; off >>= 1)
        s += __shfl_xor(s, off, 32);
    float inv  = 20.0f * rsqrtf(1e-6f + s);                   // PN_SCALE / rownorm
    const float invN = 1.0f / (float)NN;
    l1[base + lane]      = v0 * inv - csum[lane]      * invN;
    l1[base + lane + 32] = v1 * inv - csum[lane + 32] * invN;
    l1[base + lane + 64] = v2 * inv - csum[lane + 64] * invN;
}

// ================================================================ launch
extern "C" void kernel_launch(void* const* d_in, const int* in_sizes, int n_in,
                              void* d_out, int out_size, void* d_ws, size_t ws_size,
                              hipStream_t stream) {
    const float* x     = (const float*)d_in[0];
    const int*   ei    = (const int*)  d_in[1];
    const float* W0    = (const float*)d_in[2];
    const float* b0    = (const float*)d_in[3];
    const float* W1    = (const float*)d_in[4];
    const float* b1    = (const float*)d_in[5];
    const float* gamma = (const float*)d_in[6];
    const float* beta  = (const float*)d_in[7];
    const float* W2    = (const float*)d_in[8];
    const float* b2    = (const float*)d_in[9];

    float* l1  = (float*)d_out;                 // output 0: [N,H]
    float* h0  = (float*)d_out + (size_t)NN * HH; // output 1: [N,H]

    // ws layout (floats): [0, N*H) agg (later reused in-place as h_pre)
    //                     then stats: csum_h[96] csum2_h[96] scale[96] shift[96] csum_l1[96]
    float* agg     = (float*)d_ws;
    float* stats   = agg + (size_t)NN * HH;
    float* csum_h  = stats;
    float* csum2_h = stats + 96;
    float* scale   = stats + 192;
    float* shift   = stats + 288;
    float* csum_l1 = stats + 384;

    const int nzero = NN * HH + 512;            // agg + all stats
    k_zero<<<(nzero + 255) / 256, 256, 0, stream>>>(agg, nzero);

    // h0 = x @ W0^T + b0   (WMMA fp32)
    {
        int waves = RTILES, block = 128;        // 4 waves/block
        k_gemm0<<<(waves * 32 + block - 1) / block, block, 0, stream>>>(x, W0, b0, h0);
    }

    // agg[dst] += h0[src]  (77M fp32 atomics — the true bottleneck)
    {
        int total = EE * HH;
        k_scatter<<<(total + 255) / 256, 256, 0, stream>>>(h0, ei, agg);
    }

    // h = (h0 + agg) @ W1^T + b1, written in-place over agg
    {
        int waves = RTILES, block = 128;
        k_gemm1<<<(waves * 32 + block - 1) / block, block, 0, stream>>>(h0, agg, W1, b1, agg);
    }

    // BatchNorm stats -> per-feature scale/shift
    k_colstats<<<256, HH, 0, stream>>>(agg, csum_h, csum2_h);
    k_bnparams<<<1, HH, 0, stream>>>(csum_h, csum2_h, gamma, beta, scale, shift);

    // l1 = relu(bn(h)) @ W2^T + b2
    {
        int waves = RTILES, block = 128;
        k_gemm2<<<(waves * 32 + block - 1) / block, block, 0, stream>>>(agg, scale, shift, W2, b2, l1);
    }

    // PairNorm: column mean then per-row scale-shift (in place on l1)
    k_colsum<<<256, HH, 0, stream>>>(l1, csum_l1);
    {
        int waves = NN, block = 256;            // 8 waves/block, 1 row/wave
        k_pairnorm<<<(waves * 32 + block - 1) / block, block, 0, stream>>>(l1, csum_l1);
    }
}